// global_non_local_32074815767244
// MI455X (gfx1250) — compile-verified
//
#include <hip/hip_runtime.h>
#include <hip/hip_bf16.h>

typedef float v2f __attribute__((ext_vector_type(2)));
typedef float v8f __attribute__((ext_vector_type(8)));

// ---------------------------------------------------------------------------
// Kernel 1: adaptive avg pool 64x64 -> 16x16.  x:[B,256,64,64] -> xp:[B,256,256]
// One thread per pooled element; 4 float4 row loads (coalesced across pw).
// ---------------------------------------------------------------------------
__global__ void pool16_kernel(const float* __restrict__ x, float* __restrict__ xp) {
    long idx = (long)blockIdx.x * 256 + threadIdx.x;   // B*C*256 = 4,194,304
    int  n   = (int)(idx & 255);
    long bc  = idx >> 8;                               // b*256 + c
    int  ph  = n >> 4, pw = n & 15;
    const float* p = x + ((bc * 64) + ph * 4) * 64 + pw * 4;
    float s = 0.0f;
#pragma unroll
    for (int r = 0; r < 4; ++r) {
        float4 v = *(const float4*)(p + r * 64);
        s += v.x + v.y + v.z + v.w;
    }
    xp[idx] = s * (1.0f / 16.0f);
}

// ---------------------------------------------------------------------------
// Generic fp32 WMMA GEMM: one wave computes one 16x16 tile of
//   D[m,n] = sum_k A[m,k]*B[k,n]  (+ bias, + residual)
// Strides and K are compile-time so per-k addressing folds into the
// global_load immediate-offset field, and unit-K-stride operands load as
// aligned b64 (v2f).  V_WMMA_F32_16X16X4_F32 fragment layouts (ISA 7.12.2):
//   A 16x4 : lane m=l&15, regs hold K = 2*(l>>4)+{0,1}
//   B 4x16 : lane n=l&15, regs hold K = 2*(l>>4)+{0,1}
//   C/D    : lane n=l&15, reg r holds row M = r + 8*(l>>4)
// BIAS: 0=none, 1=bias[row], 2=bias[col].  HASR: add residual R (D strides).
// ---------------------------------------------------------------------------
template <long As0, long As1, long Bs0, long Bs1, long Ds0, long Ds1,
          int K, int BIAS, bool HASR>
__global__ __launch_bounds__(128)
void gemm16_wmma_kernel(const float* __restrict__ A,
                        const float* __restrict__ B,
                        const float* __restrict__ bias,
                        const float* __restrict__ R,
                        float* __restrict__ D,
                        int tilesM, int tilesN,
                        long Ab, long Bb, long Db, long Rb) {
    int wid  = blockIdx.x * 4 + (threadIdx.x >> 5);    // global wave id
    int l    = threadIdx.x & 31;
    int ln   = l & 15;                                 // m for A, n for B/D
    int hi   = l >> 4;
    int koff = hi * 2;

    int tilesPB = tilesM * tilesN;
    int batch = wid / tilesPB;
    int t     = wid - batch * tilesPB;
    int tm    = t / tilesN;
    int tn    = t - tm * tilesN;

    const float* Ap = A + (long)batch * Ab + (long)(tm * 16 + ln) * As0 + (long)koff * As1;
    const float* Bp = B + (long)batch * Bb + (long)(tn * 16 + ln) * Bs1 + (long)koff * Bs0;

    v8f acc;
    if constexpr (BIAS == 2) {
        float bv = bias[tn * 16 + ln];
#pragma unroll
        for (int r = 0; r < 8; ++r) acc[r] = bv;
    } else if constexpr (BIAS == 1) {
#pragma unroll
        for (int r = 0; r < 8; ++r) acc[r] = bias[tm * 16 + r + 8 * hi];
    } else {
#pragma unroll
        for (int r = 0; r < 8; ++r) acc[r] = 0.0f;
    }

#pragma unroll
    for (int k = 0; k < K; k += 4) {
        v2f a, b;
        if constexpr (As1 == 1) {
            a = *(const v2f*)(Ap + k);                 // aligned b64 load
        } else {
            a.x = Ap[(long)k * As1];
            a.y = Ap[(long)k * As1 + As1];
        }
        if constexpr (Bs0 == 1) {
            b = *(const v2f*)(Bp + k);                 // aligned b64 load
        } else {
            b.x = Bp[(long)k * Bs0];
            b.y = Bp[(long)k * Bs0 + Bs0];
        }
        acc = __builtin_amdgcn_wmma_f32_16x16x4_f32(
            /*neg_a=*/false, a, /*neg_b=*/false, b,
            /*c_mod=*/(short)0, acc, /*reuse_a=*/false, /*reuse_b=*/false);
    }

    long dbase = (long)batch * Db + ((long)tn * 16 + ln) * Ds1;
    long rbase = (long)batch * Rb + ((long)tn * 16 + ln) * Ds1;
#pragma unroll
    for (int r = 0; r < 8; ++r) {
        long row = tm * 16 + r + 8 * hi;
        float v  = acc[r];
        if constexpr (HASR) v += R[rbase + row * Ds0];
        D[dbase + row * Ds0] = v;
    }
}

// ---------------------------------------------------------------------------
// Softmax over rows of length 256. One wave per row (8 elems/lane).
// ---------------------------------------------------------------------------
__global__ void softmax256_kernel(float* __restrict__ S) {
    int wid = blockIdx.x * 4 + (threadIdx.x >> 5);
    int l   = threadIdx.x & 31;
    float* row = S + (long)wid * 256;
    float v[8];
    float mx = -3.4e38f;
#pragma unroll
    for (int i = 0; i < 8; ++i) { v[i] = row[l + 32 * i]; mx = fmaxf(mx, v[i]); }
#pragma unroll
    for (int off = 16; off; off >>= 1) mx = fmaxf(mx, __shfl_xor(mx, off, 32));
    float sum = 0.0f;
#pragma unroll
    for (int i = 0; i < 8; ++i) { v[i] = __expf(v[i] - mx); sum += v[i]; }
#pragma unroll
    for (int off = 16; off; off >>= 1) sum += __shfl_xor(sum, off, 32);
    float inv = 1.0f / sum;
#pragma unroll
    for (int i = 0; i < 8; ++i) row[l + 32 * i] = v[i] * inv;
}

// ---------------------------------------------------------------------------
// Nearest upsample x4: out_pooled[B,256,16,16] -> out[B,256,64,64].
// Each thread writes one splatted float4 (w group of 4 shares one src pixel).
// ---------------------------------------------------------------------------
__global__ void upsample4_kernel(const float* __restrict__ P, float* __restrict__ out) {
    long idx = (long)blockIdx.x * 256 + threadIdx.x;   // B*C*64*16 = 16,777,216
    int  t   = (int)(idx & 15);
    int  h   = (int)((idx >> 4) & 63);
    long bc  = idx >> 10;                              // b*256 + c
    float v  = P[bc * 256 + (h >> 2) * 16 + t];
    float4 o = make_float4(v, v, v, v);
    *(float4*)(out + (bc * 64 + h) * 64 + 4 * t) = o;
}

// ---------------------------------------------------------------------------
extern "C" void kernel_launch(void* const* d_in, const int* in_sizes, int n_in,
                              void* d_out, int out_size, void* d_ws, size_t ws_size,
                              hipStream_t stream) {
    const float* x       = (const float*)d_in[0];
    const float* theta_w = (const float*)d_in[1];
    const float* theta_b = (const float*)d_in[2];
    const float* phi_w   = (const float*)d_in[3];
    const float* phi_b   = (const float*)d_in[4];
    const float* g_w     = (const float*)d_in[5];
    const float* g_b     = (const float*)d_in[6];
    const float* out_w   = (const float*)d_in[7];
    const float* out_b   = (const float*)d_in[8];
    float* out = (float*)d_out;

    // Scratch layout (floats). Total = 14,680,064 floats = 56 MB (fits in L2).
    float* ws    = (float*)d_ws;
    float* xp    = ws;                    // [64][256][256]  16 MB
    float* theta = xp    + 4194304;       // [64][256][128]   8 MB (reused as agg)
    float* phi   = theta + 2097152;       // [64][128][256]   8 MB
    float* g     = phi   + 2097152;       // [64][256][128]   8 MB
    float* attn  = g     + 2097152;       // [64][256][256]  16 MB (reused as out_pooled)
    float* agg   = theta;                 // overlay: theta dead after scores
    float* outp  = attn;                  // overlay: attn dead after agg

    // 1) pool: x -> xp[b][c][n]
    pool16_kernel<<<16384, 256, 0, stream>>>(x, xp);

    // 2) theta_t[b][n][o] = sum_c xp[b][c][n]*theta_w[o][c] + theta_b[o]
    gemm16_wmma_kernel<1, 256, 1, 256, 128, 1, 256, 2, false>
        <<<2048, 128, 0, stream>>>(xp, theta_w, theta_b, nullptr, theta,
                                   16, 8, 65536, 0, 32768, 0);

    // 3) phi[b][o][n] = sum_c phi_w[o][c]*xp[b][c][n] + phi_b[o]
    gemm16_wmma_kernel<256, 1, 256, 1, 256, 1, 256, 1, false>
        <<<2048, 128, 0, stream>>>(phi_w, xp, phi_b, nullptr, phi,
                                   8, 16, 0, 65536, 32768, 0);

    // 4) g_t[b][n][o] = sum_c xp[b][c][n]*g_w[o][c] + g_b[o]
    gemm16_wmma_kernel<1, 256, 1, 256, 128, 1, 256, 2, false>
        <<<2048, 128, 0, stream>>>(xp, g_w, g_b, nullptr, g,
                                   16, 8, 65536, 0, 32768, 0);

    // 5) scores[b][i][j] = sum_o theta_t[b][i][o]*phi[b][o][j]
    gemm16_wmma_kernel<128, 1, 256, 1, 256, 1, 128, 0, false>
        <<<4096, 128, 0, stream>>>(theta, phi, nullptr, nullptr, attn,
                                   16, 16, 32768, 32768, 65536, 0);

    // 6) softmax over last dim of attn
    softmax256_kernel<<<4096, 128, 0, stream>>>(attn);

    // 7) agg_t[b][n][o] = sum_j attn[b][n][j]*g_t[b][j][o]   (agg overlays theta)
    gemm16_wmma_kernel<256, 1, 128, 1, 128, 1, 256, 0, false>
        <<<2048, 128, 0, stream>>>(attn, g, nullptr, nullptr, agg,
                                   16, 8, 65536, 32768, 32768, 0);

    // 8) out_pooled[b][co][n] = sum_o out_w[co][o]*agg_t[b][n][o] + out_b[co] + xp[b][co][n]
    //    (out_pooled overlays attn)
    gemm16_wmma_kernel<128, 1, 1, 128, 256, 1, 128, 1, true>
        <<<4096, 128, 0, stream>>>(out_w, agg, out_b, xp, outp,
                                   16, 16, 0, 32768, 65536, 65536);

    // 9) nearest x4 upsample to [64,256,64,64]
    upsample4_kernel<<<65536, 256, 0, stream>>>(outp, out);
}